// SoftRankTransform_89421219103237
// MI455X (gfx1250) — compile-verified
//
#include <hip/hip_runtime.h>
#include <float.h>

typedef __attribute__((ext_vector_type(2))) float        v2f;
typedef __attribute__((ext_vector_type(8))) float        v8f;
typedef __attribute__((ext_vector_type(4))) unsigned int v4u;
typedef __attribute__((ext_vector_type(8))) int          v8i;
typedef __attribute__((ext_vector_type(4))) int          v4i;

#define NCOL 160
#define NIDX 128
#define MS   129              // padded row stride (words) for the minimax matrix
#define INV_REG   10.0f       // 1 / regularization_strength
#define OUT_SCALE 0.0078125f  // 1 / DIM

// One block (128 threads = 4 waves, wave32) per row.
__global__ __launch_bounds__(NIDX) void softrank_kernel(const float* __restrict__ X,
                                                        const int* __restrict__ indices,
                                                        float* __restrict__ out) {
  __shared__ float rowbuf[NCOL];        // raw row staged by the TDM
  __shared__ float sval[NIDX];          // theta, then sorted-descending theta
  __shared__ int   sidx[NIDX];          // permutation (original column per rank)
  __shared__ float psum[NIDX];          // inclusive prefix sums of (w - s)
  __shared__ float rlen[NIDX];          // rlen[L] = 1 / (L+1)
  __shared__ unsigned char used[NCOL];  // membership mask of `indices`
  extern __shared__ float M[];          // NIDX x MS seg-mean / cummin matrix

  const int tid = threadIdx.x;
  const int row = blockIdx.x;
  const float* rp = X + (size_t)row * NCOL;

  for (int t = tid; t < NCOL; t += NIDX) used[t] = 0;
  rlen[tid] = 1.0f / (float)(tid + 1);   // one exact division per thread

#if __has_builtin(__builtin_amdgcn_tensor_load_to_lds)
  // ---- Tensor Data Mover: DMA the whole 160-float row into LDS (wave 0) ----
  if (tid < 32) {
    unsigned long long ga = (unsigned long long)(size_t)rp;
    unsigned int lds = (unsigned int)(size_t)&rowbuf[0];
    v4u g0;
    g0.x = 1u;                                                     // count = 1
    g0.y = lds;                                                    // lds_addr
    g0.z = (unsigned int)ga;                                       // global_addr[31:0]
    g0.w = (unsigned int)((ga >> 32) & 0x1FFFFFFull) | (2u << 30); // addr[56:32] | type=2
    v8i g1;
    g1[0] = (int)(2u << 16);             // data_size = 4 bytes
    g1[1] = (int)((unsigned)NCOL << 16); // tensor_dim0 = 160 (bits 79:48)
    g1[2] = (int)(1u << 16);             // tensor_dim1 = 1   (bits 111:80)
    g1[3] = (int)((unsigned)NCOL << 16); // tile_dim0   = 160 (bits 127:112)
    g1[4] = 1;                           // tile_dim1   = 1   (bits 143:128)
    g1[5] = NCOL;                        // tensor_dim0_stride = 160
    g1[6] = 0;
    g1[7] = 0;
    v4i gz4 = {0, 0, 0, 0};
    v8i gz8 = {0, 0, 0, 0, 0, 0, 0, 0};
    __builtin_amdgcn_tensor_load_to_lds(g0, g1, gz4, gz4, gz8, 0);
    __builtin_amdgcn_s_wait_tensorcnt(0);
  }
#else
  for (int t = tid; t < NCOL; t += NIDX) rowbuf[t] = rp[t];
#endif
  __syncthreads();

  {
    const int c = indices[tid];
    sval[tid] = rowbuf[c] * INV_REG;   // theta = values / reg_strength
    sidx[tid] = c;
    used[c] = 1;
  }

  // ---- bitonic sort of 128 keys: (value descending, original index ascending)
  for (int k = 2; k <= NIDX; k <<= 1) {
    for (int d = k >> 1; d > 0; d >>= 1) {
      __syncthreads();
      if ((tid & d) == 0) {
        const int p = tid | d;
        const bool up = ((tid & k) == 0);  // final stage: whole array rank-ascending
        float va = sval[tid], vb = sval[p];
        int   ia = sidx[tid], ib = sidx[p];
        const bool aFirst = (va > vb) || (va == vb && ia < ib);
        if (aFirst != up) {
          sval[tid] = vb; sval[p] = va;
          sidx[tid] = ib; sidx[p] = ia;
        }
      }
    }
  }
  __syncthreads();

  // ---- isotonic input x_i = w_i - s_i (w_i = NIDX - i); inclusive prefix scan
  psum[tid] = (float)(NIDX - tid) - sval[tid];
  __syncthreads();
  for (int off = 1; off < NIDX; off <<= 1) {
    float t = psum[tid] + ((tid >= off) ? psum[tid - off] : 0.0f);
    __syncthreads();
    psum[tid] = t;
    __syncthreads();
  }

  // ---- seg-mean matrix S[j,k] = (P[k+1]-P[j])/(k-j+1) via rank-2 WMMA tiles:
  //      S_tile(16x16) = A(16x4) x B(4x16), A=[-Pexcl, 1, 0, 0], B=[1; Pincl; 0; 0]
  const int wave  = tid >> 5;
  const int lane  = tid & 31;
  const int laneN = lane & 15;
  const int hi    = lane >> 4;
#if __has_builtin(__builtin_amdgcn_wmma_f32_16x16x4_f32)
  for (int t = 0; t < 16; ++t) {
    const int tile = wave * 16 + t;
    const int j0 = (tile >> 3) * 16;
    const int k0 = (tile & 7) * 16;
    const float Pexcl = (j0 + laneN == 0) ? 0.0f : psum[j0 + laneN - 1];
    v2f a, b;
    a.x = hi ? 0.0f : -Pexcl;            // A VGPR0: K=0 (lanes0-15) / K=2 (=0)
    a.y = hi ? 0.0f : 1.0f;              // A VGPR1: K=1            / K=3 (=0)
    b.x = hi ? 0.0f : 1.0f;              // B VGPR0: row K=0 = ones / K=2 (=0)
    b.y = hi ? 0.0f : psum[k0 + laneN];  // B VGPR1: row K=1 = Pincl/ K=3 (=0)
    v8f c = {0.f, 0.f, 0.f, 0.f, 0.f, 0.f, 0.f, 0.f};
    c = __builtin_amdgcn_wmma_f32_16x16x4_f32(false, a, false, b, (short)0, c,
                                              false, false);
    const int kk = k0 + laneN;
#pragma unroll
    for (int r = 0; r < 8; ++r) {
      const int j = j0 + r + 8 * hi;     // C/D layout: VGPR r -> M=r (+8 upper half)
      const int L = kk - j;              // segment length - 1
      const float m = c[r] * rlen[L & (NIDX - 1)];  // branchless: masked index
      M[j * MS + kk] = (L >= 0) ? m : FLT_MAX;
    }
  }
#else
  for (int e = tid; e < NIDX * NIDX; e += NIDX) {
    const int j = e >> 7, kk = e & (NIDX - 1);
    const float Pex = (j == 0) ? 0.0f : psum[j - 1];
    const int L = kk - j;
    const float m = (psum[kk] - Pex) * rlen[L & (NIDX - 1)];
    M[j * MS + kk] = (L >= 0) ? m : FLT_MAX;
  }
#endif
  __syncthreads();

  // ---- reverse cummin along k (thread owns row j = tid)
  {
    float run = M[tid * MS + (NIDX - 1)];
    for (int kk = NIDX - 2; kk >= 0; --kk) {
      run = fminf(run, M[tid * MS + kk]);
      M[tid * MS + kk] = run;
    }
  }
  __syncthreads();

  // ---- y_i = max_{j<=i} cmin[j,i]  (diagonal of the cummax)
  float y = M[tid];                           // j = 0, column i = tid
  for (int j = 1; j <= tid; ++j) y = fmaxf(y, M[j * MS + tid]);

  // dual = -y, primal = s - dual = s + y; scatter through the permutation
  out[(size_t)row * NCOL + sidx[tid]] = (sval[tid] + y) * OUT_SCALE;

  // pass-through columns not covered by `indices`
  for (int t = tid; t < NCOL; t += NIDX)
    if (!used[t]) out[(size_t)row * NCOL + t] = rowbuf[t] * OUT_SCALE;
}

extern "C" void kernel_launch(void* const* d_in, const int* in_sizes, int n_in,
                              void* d_out, int out_size, void* d_ws, size_t ws_size,
                              hipStream_t stream) {
  (void)n_in; (void)out_size; (void)d_ws; (void)ws_size;
  const float* X       = (const float*)d_in[0];
  const int*   indices = (const int*)d_in[1];
  float*       out     = (float*)d_out;
  const int rows = in_sizes[0] / NCOL;                       // 8192
  const size_t shbytes = (size_t)NIDX * MS * sizeof(float);  // ~66 KB dynamic LDS
  softrank_kernel<<<rows, NIDX, shbytes, stream>>>(X, indices, out);
}